// IDCTConvLayer_39049842655608
// MI455X (gfx1250) — compile-verified
//
#include <hip/hip_runtime.h>

typedef float v2f __attribute__((ext_vector_type(2)));
typedef float v8f __attribute__((ext_vector_type(8)));

// M[n][k] = 2*cos(pi*(2n+1)*k/14), with column k=0 forced to 1.0 (matches reference).
__device__ const float MTAB[49] = {
  1.0f,  1.9498558244f,  1.8019377358f,  1.5636629649f,  1.2469796037f,  0.8677674782f,  0.4450418679f,
  1.0f,  1.5636629649f,  0.4450418679f, -0.8677674782f, -1.8019377358f, -1.9498558244f, -1.2469796037f,
  1.0f,  0.8677674782f, -1.2469796037f, -1.9498558244f, -0.4450418679f,  1.5636629649f,  1.8019377358f,
  1.0f,  0.0f,          -2.0f,           0.0f,           2.0f,           0.0f,          -2.0f,
  1.0f, -0.8677674782f, -1.2469796037f,  1.9498558244f, -0.4450418679f, -1.5636629649f,  1.8019377358f,
  1.0f, -1.5636629649f,  0.4450418679f,  0.8677674782f, -1.8019377358f,  1.9498558244f, -1.2469796037f,
  1.0f, -1.9498558244f,  1.8019377358f, -1.5636629649f,  1.2469796037f, -0.8677674782f,  0.4450418679f,
};

#define HH 384
#define WW 384
#define CH 48
#define TILES_X 24          // 384 / 16
#define WAVES_PER_BLK 8

__global__ __launch_bounds__(256)
void idct_wmma_kernel(const float* __restrict__ X,
                      const float* __restrict__ zf,
                      float* __restrict__ out)
{
    __shared__ float mt[49];
    __shared__ float stage[WAVES_PER_BLK][256];

    const int tid = threadIdx.x;
    if (tid < 49) mt[tid] = MTAB[tid];
    __syncthreads();

    const int wave = tid >> 5;
    const int lane = tid & 31;
    const int hi   = lane >> 4;      // which lane-half (K split)
    const int lo   = lane & 15;      // M (pixel) for A, N (column) for B/D

    const int tile = blockIdx.x * WAVES_PER_BLK + wave;      // 0 .. 73727
    const int b    = tile / (HH * TILES_X);
    const int rem  = tile - b * (HH * TILES_X);
    const int y    = rem / TILES_X;
    const int tx0  = (rem - y * TILES_X) * 16;

    const int i  = (y < 377) ? y : 376;          // clamped row
    int dy = y - i;  if (dy > 6) dy = 6;         // dy==7 only at y==383 (masked out)

    // ---- A: 16 pixels x 48 channels, per documented f32 A-matrix layout ----
    const int xp = tx0 + lo;
    const int j  = (xp < 377) ? xp : 376;
    const float* aptr = X + ((size_t)((b * HH + i) * WW + j) * CH + 2 * hi);

    v2f a[12];
#pragma unroll
    for (int s = 0; s < 12; ++s)
        a[s] = *(const v2f*)(aptr + 4 * s);      // global_load_b64, coalesced

    // ---- B: columns 0..6 = weight vectors for dx=0..6, columns 7..15 = 0 ----
    float Ar[7], Bc[7];
#pragma unroll
    for (int u = 0; u < 7; ++u) Ar[u] = mt[dy * 7 + u];
    const int bcol = (lo < 7) ? lo : 0;
#pragma unroll
    for (int v = 0; v < 7; ++v) Bc[v] = mt[bcol * 7 + v];
    const float cmask = (lo < 7) ? 1.0f : 0.0f;

    v2f bm[12];
    if (hi == 0) {
#pragma unroll
        for (int s = 0; s < 12; ++s) {
            const int k0 = 4 * s + 1;            // coeff flat index (u,v)=divmod(k,7)
            const int k1 = 4 * s + 2;
            bm[s].x = Ar[k0 / 7] * Bc[k0 % 7] * cmask;
            bm[s].y = Ar[k1 / 7] * Bc[k1 % 7] * cmask;
        }
    } else {
#pragma unroll
        for (int s = 0; s < 12; ++s) {
            const int k0 = 4 * s + 3;
            const int k1 = 4 * s + 4;
            bm[s].x = Ar[k0 / 7] * Bc[k0 % 7] * cmask;
            bm[s].y = Ar[k1 / 7] * Bc[k1 % 7] * cmask;
        }
    }

    // ---- 12 chained V_WMMA_F32_16X16X4_F32: D[pixel, dx-case] ----
    v8f acc = {};
#pragma unroll
    for (int s = 0; s < 12; ++s)
        acc = __builtin_amdgcn_wmma_f32_16x16x4_f32(
            /*neg_a=*/false, a[s], /*neg_b=*/false, bm[s],
            /*c_mod=*/(short)0, acc, /*reuse_a=*/false, /*reuse_b=*/false);

    // ---- stage D through LDS: st[N*16 + M] ----
    float* st = &stage[wave][0];
#pragma unroll
    for (int r = 0; r < 8; ++r)
        st[lo * 16 + 8 * hi + r] = acc[r];       // M = r + 8*hi, N = lo
    __syncthreads();

    // ---- per-pixel select + zero-freq add + boundary mask, coalesced store ----
    if (lane < 16) {
        const int x  = tx0 + lane;
        const int jj = (x < 377) ? x : 376;
        int dx = x - jj;  if (dx > 6) dx = 6;    // dx==7 only at x==383 (masked)
        float v = st[dx * 16 + lane];
        float z = zf[(size_t)(b * HH + i) * WW + jj];
        float res = z + v;
        if (y == HH - 1 || x == WW - 1) res = 0.0f;
        out[(size_t)(b * HH + y) * WW + x] = res;
    }
}

extern "C" void kernel_launch(void* const* d_in, const int* in_sizes, int n_in,
                              void* d_out, int out_size, void* d_ws, size_t ws_size,
                              hipStream_t stream) {
    const float* X  = (const float*)d_in[0];   // (8,384,384,48) f32
    const float* zf = (const float*)d_in[1];   // (8,384,384,1)  f32
    float* out = (float*)d_out;                // (8,384,384,1)  f32

    const int total_tiles = 8 * HH * TILES_X;          // 73728 waves
    const int blocks = total_tiles / WAVES_PER_BLK;    // 9216
    idct_wmma_kernel<<<blocks, 256, 0, stream>>>(X, zf, out);
}